// FiringRateModel_17643725652341
// MI455X (gfx1250) — compile-verified
//
#include <hip/hip_runtime.h>

// FiringRateModel recurrence for MI455X (gfx1250, wave32).
//
// Design: one wave = 16 independent batch chains. State u[n]=v[n]*w[n] for
// n=0..63 lives in 16 v2f registers laid out as V_WMMA_F32_16X16X4_F32
// B-operands (columns = batch = lane&15). Per time step:
//   - 96 VALU ops update u (v_fma chains, per-lane constants in VGPRs)
//   - 16 v_wmma_f32_16x16x4_f32 with an all-ones A matrix reduce z = sum_n u[n]
//     on the matrix pipe (co-executes with the VALU update). All rows of D are
//     identical, so D[0] holds z[batch] duplicated in both lane halves -- the
//     exact broadcast layout needed to feed f back into the next update.
//   - Horner poly + tanh (v_exp_f32 + v_rcp_f32) + relu, store lanes 0..15.
// currents[t+1] is register-prefetched to hide the single global load.

static constexpr int T_STEPS = 2048;
static constexpr int BATCH   = 8192;
static constexpr int NNEUR   = 64;

typedef __attribute__((ext_vector_type(2))) float v2f;
typedef __attribute__((ext_vector_type(8))) float v8f;

__global__ __launch_bounds__(128) void firing_rate_kernel(
    const float* __restrict__ currents,   // [T, B]
    const float* __restrict__ fs0,        // [B]
    const float* __restrict__ a,          // [N]
    const float* __restrict__ b,          // [N]
    const float* __restrict__ w,          // [N, 1]
    const float* __restrict__ ds,         // [N]
    const float* __restrict__ pc,         // [DEG+1]
    const float* __restrict__ g_b,        // scalar
    float* __restrict__ out)              // [T, B]
{
    const int lane   = threadIdx.x & 31;
    const int waveId = blockIdx.x * (blockDim.x >> 5) + (threadIdx.x >> 5);
    const int col    = lane & 15;            // batch within wave
    const int h      = lane >> 4;            // lane half (0/1)
    const int batch  = (waveId << 4) + col;

    // ---- per-lane constants in WMMA-B layout: n = 4c + 2e + h ------------
    v2f u[16], q[16], aw[16], bw[16];
#pragma unroll
    for (int c = 0; c < 16; ++c) {
        const int n0 = 4 * c + h;
        const int n1 = 4 * c + 2 + h;
        const float w0 = w[n0], w1 = w[n1];
        aw[c] = (v2f){ a[n0] * w0,            a[n1] * w1 };
        bw[c] = (v2f){ 1000.0f * b[n0] * w0,  1000.0f * b[n1] * w1 };
        q[c]  = (v2f){ 1.0f - ds[n0],         1.0f - ds[n1] };
        u[c]  = (v2f){ 0.0f, 0.0f };          // v0 == 0
    }

    // squared polynomial coefficients + bias
    const float c0 = pc[0] * pc[0], c1 = pc[1] * pc[1], c2 = pc[2] * pc[2];
    const float c3 = pc[3] * pc[3], c4 = pc[4] * pc[4], c5 = pc[5] * pc[5];
    const float gb = g_b[0];
    const float inv_max_i = 1.0f / 100.0f;   // 1/MAX_I

    const v2f ones = (v2f){ 1.0f, 1.0f };

    float f   = fs0[batch];
    float cur = currents[batch];             // t = 0, prefetched

    for (int t = 0; t < T_STEPS; ++t) {
        // software prefetch of next step's current (hides load latency)
        const int tn = (t + 1 < T_STEPS) ? (t + 1) : t;
        const float cur_next = currents[(size_t)tn * BATCH + batch];

        const v2f cv = (v2f){ cur, cur };
        const v2f fv = (v2f){ f,   f   };

        v8f acc0 = {};
        v8f acc1 = {};
#pragma unroll
        for (int c = 0; c < 16; ++c) {
            // u = q*u + (cur*aw + f*bw)   -- 3 VALU per v2f element pair
            u[c] = u[c] * q[c] + (cv * aw[c] + fv * bw[c]);
            // matrix-pipe reduction: D = ones(16x4) x u-chunk(4x16) + C
            // two interleaved accumulators halve the serial WMMA chain
            if (c & 1)
                acc1 = __builtin_amdgcn_wmma_f32_16x16x4_f32(
                    false, ones, false, u[c], (short)0, acc1, false, false);
            else
                acc0 = __builtin_amdgcn_wmma_f32_16x16x4_f32(
                    false, ones, false, u[c], (short)0, acc0, false, false);
        }

        // every row of D is z; element 0 carries z[batch] in both lane halves
        const float z = acc0[0] + acc1[0];

        // polynomial activation: relu(MAX_FR * tanh(poly((z-gb)/MAX_I)))
        const float x = (z - gb) * inv_max_i;
        float p = c5;
        p = p * x + c4;
        p = p * x + c3;
        p = p * x + c2;
        p = p * x + c1;
        p = p * x + c0;

        // tanh(p) = 1 - 2/(exp(2p)+1)  via v_exp_f32 + v_rcp_f32
        const float e  = __expf(2.0f * p);
        const float th = 1.0f - 2.0f * __builtin_amdgcn_rcpf(e + 1.0f);
        float fr = 200.0f * th;
        fr = fr > 0.0f ? fr : 0.0f;
        f = fr;

        if (h == 0)                           // lanes 0..15 own the 16 batches
            out[(size_t)t * BATCH + batch] = fr;

        cur = cur_next;
    }
}

extern "C" void kernel_launch(void* const* d_in, const int* in_sizes, int n_in,
                              void* d_out, int out_size, void* d_ws, size_t ws_size,
                              hipStream_t stream) {
    (void)in_sizes; (void)n_in; (void)out_size; (void)d_ws; (void)ws_size;
    const float* currents = (const float*)d_in[0];
    const float* fs0      = (const float*)d_in[1];
    const float* a        = (const float*)d_in[2];
    const float* b        = (const float*)d_in[3];
    const float* w        = (const float*)d_in[4];
    const float* ds       = (const float*)d_in[5];
    const float* pc       = (const float*)d_in[6];
    const float* g_b      = (const float*)d_in[7];
    float* out = (float*)d_out;

    // 8192 batches / 16 per wave = 512 waves; 4 waves (128 threads) per block
    dim3 block(128);
    dim3 grid(BATCH / (16 * 4));
    firing_rate_kernel<<<grid, block, 0, stream>>>(
        currents, fs0, a, b, w, ds, pc, g_b, out);
}